// SMPL_85779086835871
// MI455X (gfx1250) — compile-verified
//
#include <hip/hip_runtime.h>
#include <hip/hip_bf16.h>

typedef float v2f __attribute__((ext_vector_type(2)));
typedef float v8f __attribute__((ext_vector_type(8)));

#define N_VERT 100000
#define N_ROWS 300000        // N_VERT * 3
#define K_SHAPE 300
#define K_POSE 207
#define N_TILES_ROWS 18750   // N_ROWS / 16
#define N_TILES_VERT 6250    // N_VERT / 16

// ---------------------------------------------------------------------------
// Kernel 1: v_shaped = shapedirs(300000x300) @ beta(300) + v_template
// One wave per 16-row tile; WMMA f32 16x16x4, beta broadcast across N.
// ---------------------------------------------------------------------------
__global__ void __launch_bounds__(256) shape_blend_kernel(
    const float* __restrict__ shapedirs,   // [300000 x 300]
    const float* __restrict__ beta,        // [300]
    const float* __restrict__ v_template,  // [300000]
    float* __restrict__ v_shaped)          // [300000]
{
  __shared__ float beta_s[304];
  const int tid = threadIdx.x;
  for (int i = tid; i < K_SHAPE; i += 256) beta_s[i] = beta[i];
  __syncthreads();

  const int lane = tid & 31;
  const int half = lane >> 4;   // 0: k pair {0,1}; 1: k pair {2,3}
  const int m    = lane & 15;   // A row within tile
  const long tile = (long)blockIdx.x * 8 + (tid >> 5);
  if (tile >= N_TILES_ROWS) return;
  const long r0 = tile * 16;
  const float* arow = shapedirs + (r0 + m) * (long)K_SHAPE + 2 * half;

  v8f c = {};
  for (int kb = 0; kb < K_SHAPE; kb += 4) {
    v2f a = *(const v2f*)(arow + kb);                 // 8B aligned (even offsets)
    v2f b;
    b.x = beta_s[kb + 2 * half];
    b.y = beta_s[kb + 2 * half + 1];
    c = __builtin_amdgcn_wmma_f32_16x16x4_f32(false, a, false, b,
                                              (short)0, c, false, false);
  }
  // every column n holds identical values; lane n==0 of each half writes rows
  if (m == 0) {
    const long base = r0 + 8 * half;
#pragma unroll
    for (int r = 0; r < 8; ++r)
      v_shaped[base + r] = c[r] + v_template[base + r];
  }
}

// ---------------------------------------------------------------------------
// Kernel 2: J[j,d] = sum_v J_regressor[j,v] * v_shaped[v,d]  (72 blocks)
// ---------------------------------------------------------------------------
__global__ void __launch_bounds__(256) joint_reduce_kernel(
    const float* __restrict__ J_regressor,  // [24 x 100000]
    const float* __restrict__ v_shaped,     // [100000 x 3]
    float* __restrict__ Jout)               // [24 x 3]
{
  __shared__ float red[256];
  const int j = blockIdx.x / 3, d = blockIdx.x % 3;
  const float* jr = J_regressor + (long)j * N_VERT;
  float acc = 0.f;
  for (int v = threadIdx.x; v < N_VERT; v += 256)
    acc += jr[v] * v_shaped[(long)v * 3 + d];
  red[threadIdx.x] = acc;
  __syncthreads();
  for (int s = 128; s > 0; s >>= 1) {
    if (threadIdx.x < s) red[threadIdx.x] += red[threadIdx.x + s];
    __syncthreads();
  }
  if (threadIdx.x == 0) Jout[j * 3 + d] = red[0];
}

// ---------------------------------------------------------------------------
// Kernel 3: Rodrigues, lrotmin (padded to 208), kinematic chain G'
// single block of 32 threads
// ---------------------------------------------------------------------------
__global__ void pose_kernel(const float* __restrict__ pose,  // [72]
                            const float* __restrict__ Jin,   // [24 x 3]
                            float* __restrict__ lrot,        // [208] (207 + pad0)
                            float* __restrict__ Gout)        // [24 x 16]
{
  __shared__ float R[24][9];
  const int i = threadIdx.x;
  if (i < 24) {
    float rx = pose[3 * i], ry = pose[3 * i + 1], rz = pose[3 * i + 2];
    float th = sqrtf(rx * rx + ry * ry + rz * rz + 1e-16f);
    float x = rx / th, y = ry / th, z = rz / th;
    float cc = cosf(th), ss = sinf(th), ic = 1.f - cc;
    R[i][0] = cc + ic * x * x;  R[i][1] = ic * x * y - ss * z;  R[i][2] = ic * x * z + ss * y;
    R[i][3] = ic * y * x + ss * z;  R[i][4] = cc + ic * y * y;  R[i][5] = ic * y * z - ss * x;
    R[i][6] = ic * z * x - ss * y;  R[i][7] = ic * z * y + ss * x;  R[i][8] = cc + ic * z * z;
  }
  __syncthreads();
  if (i >= 1 && i < 24) {
    const float I9[9] = {1, 0, 0, 0, 1, 0, 0, 0, 1};
#pragma unroll
    for (int a = 0; a < 9; ++a) lrot[(i - 1) * 9 + a] = R[i][a] - I9[a];
  }
  if (i == 0) {
    lrot[207] = 0.f;  // zero pad so the padded GEMV tail contributes nothing
    const int par[24] = {0, 0, 0, 0, 1, 2, 3, 4, 5, 6, 7, 8,
                         9, 9, 9, 12, 13, 14, 16, 17, 18, 19, 20, 21};
    float G[24][16];
    for (int a = 0; a < 3; ++a) {
      for (int b = 0; b < 3; ++b) G[0][a * 4 + b] = R[0][a * 3 + b];
      G[0][a * 4 + 3] = Jin[a];
    }
    G[0][12] = 0.f; G[0][13] = 0.f; G[0][14] = 0.f; G[0][15] = 1.f;
    for (int jj = 1; jj < 24; ++jj) {
      const int p = par[jj];
      float A[16];
      for (int a = 0; a < 3; ++a) {
        for (int b = 0; b < 3; ++b) A[a * 4 + b] = R[jj][a * 3 + b];
        A[a * 4 + 3] = Jin[jj * 3 + a] - Jin[p * 3 + a];
      }
      A[12] = 0.f; A[13] = 0.f; A[14] = 0.f; A[15] = 1.f;
      for (int a = 0; a < 4; ++a)
        for (int b = 0; b < 4; ++b) {
          float acc = 0.f;
          for (int k = 0; k < 4; ++k) acc += G[p][a * 4 + k] * A[k * 4 + b];
          G[jj][a * 4 + b] = acc;
        }
    }
    // G' : subtract G[:, :3] @ J from translation column
    for (int jj = 0; jj < 24; ++jj) {
      for (int a = 0; a < 4; ++a) {
        float rj = G[jj][a * 4 + 0] * Jin[jj * 3 + 0] +
                   G[jj][a * 4 + 1] * Jin[jj * 3 + 1] +
                   G[jj][a * 4 + 2] * Jin[jj * 3 + 2];
        G[jj][a * 4 + 3] -= rj;
      }
      for (int e = 0; e < 16; ++e) Gout[jj * 16 + e] = G[jj][e];
    }
  }
}

// ---------------------------------------------------------------------------
// Kernel 4: v_posed = v_shaped + posedirs(300000x207) @ lrotmin
// same WMMA-GEMV; K padded 207 -> 208 with lrot[207] == 0
// ---------------------------------------------------------------------------
__global__ void __launch_bounds__(256) pose_blend_kernel(
    const float* __restrict__ posedirs,  // [300000 x 207]
    const float* __restrict__ lrot,      // [208]
    const float* __restrict__ v_shaped,  // [300000]
    float* __restrict__ v_posed)         // [300000]
{
  __shared__ float b_s[208];
  const int tid = threadIdx.x;
  for (int i = tid; i < 208; i += 256) b_s[i] = lrot[i];
  __syncthreads();

  const int lane = tid & 31;
  const int half = lane >> 4;
  const int m    = lane & 15;
  const long tile = (long)blockIdx.x * 8 + (tid >> 5);
  if (tile >= N_TILES_ROWS) return;
  const long r0 = tile * 16;
  const float* arow = posedirs + (r0 + m) * (long)K_POSE;

  v8f c = {};
  // main body: k = 0..203 (rows are 207 long -> no 8B alignment guarantee,
  // use scalar b32 loads and let the compiler combine when it can)
  for (int kb = 0; kb < 204; kb += 4) {
    const int k0 = kb + 2 * half;
    v2f a, b;
    a.x = arow[k0];
    a.y = arow[k0 + 1];
    b.x = b_s[k0];
    b.y = b_s[k0 + 1];
    c = __builtin_amdgcn_wmma_f32_16x16x4_f32(false, a, false, b,
                                              (short)0, c, false, false);
  }
  // tail: k = 204..207 ; k==207 is padding (b==0), keep loads in-bounds
  {
    const int k0 = 204 + 2 * half;
    v2f a, b;
    a.x = arow[k0];                              // 204 or 206: in range
    a.y = (k0 + 1 <= 206) ? arow[k0 + 1] : 0.f;  // lane-level select only
    b.x = b_s[k0];
    b.y = b_s[k0 + 1];                            // b_s[207] == 0
    c = __builtin_amdgcn_wmma_f32_16x16x4_f32(false, a, false, b,
                                              (short)0, c, false, false);
  }
  if (m == 0) {
    const long base = r0 + 8 * half;
#pragma unroll
    for (int r = 0; r < 8; ++r)
      v_posed[base + r] = c[r] + v_shaped[base + r];
  }
}

// ---------------------------------------------------------------------------
// Kernel 5: skinning. Per 16-vertex tile:
//   D[m,n] = sum_k G'[k][m] * w[n][k]   (M=16 transform entries, K=24, N=16)
// then v[n] = T[n](3x4) . [v_posed[n], 1]
// ---------------------------------------------------------------------------
__global__ void __launch_bounds__(256) skin_kernel(
    const float* __restrict__ skin_weights,  // [100000 x 24]
    const float* __restrict__ Gm,            // [24 x 16]
    const float* __restrict__ v_posed,       // [100000 x 3]
    float* __restrict__ out)                 // [100000 x 3]
{
  __shared__ float G_s[24][16];
  const int tid = threadIdx.x;
  for (int i = tid; i < 384; i += 256) G_s[i >> 4][i & 15] = Gm[i];
  __syncthreads();

  const int lane = tid & 31;
  const int half = lane >> 4;
  const int n    = lane & 15;   // A row (transform entry) AND B column (vertex)
  const long tile = (long)blockIdx.x * 8 + (tid >> 5);
  if (tile >= N_TILES_VERT) return;
  const long vtx = tile * 16 + n;
  const float* wrow = skin_weights + vtx * 24;

  v8f c = {};
#pragma unroll
  for (int t = 0; t < 6; ++t) {
    const int k0 = 4 * t + 2 * half;
    v2f a, b;
    a.x = G_s[k0][n];        // A[m][k0]   = G'[k0][m], m = lane&15
    a.y = G_s[k0 + 1][n];
    b = *(const v2f*)(wrow + k0);  // B[k][n] = w[vtx][k]; 8B aligned
    c = __builtin_amdgcn_wmma_f32_16x16x4_f32(false, a, false, b,
                                              (short)0, c, false, false);
  }

  // c: lane holds T entries m = r + 8*half (r = vgpr idx) of vertex `vtx`
  const float px = v_posed[vtx * 3 + 0];
  const float py = v_posed[vtx * 3 + 1];
  const float pz = v_posed[vtx * 3 + 2];
  if (half == 0) {
    out[vtx * 3 + 0] = c[0] * px + c[1] * py + c[2] * pz + c[3];
    out[vtx * 3 + 1] = c[4] * px + c[5] * py + c[6] * pz + c[7];
  } else {
    out[vtx * 3 + 2] = c[0] * px + c[1] * py + c[2] * pz + c[3];
  }
}

// ---------------------------------------------------------------------------
extern "C" void kernel_launch(void* const* d_in, const int* in_sizes, int n_in,
                              void* d_out, int out_size, void* d_ws, size_t ws_size,
                              hipStream_t stream) {
  const float* beta        = (const float*)d_in[0];  // [300]
  const float* pose        = (const float*)d_in[1];  // [72]
  const float* v_template  = (const float*)d_in[2];  // [100000 x 3]
  const float* J_regressor = (const float*)d_in[3];  // [24 x 100000]
  const float* skin_w      = (const float*)d_in[4];  // [100000 x 24]
  const float* shapedirs   = (const float*)d_in[5];  // [100000 x 3 x 300]
  const float* posedirs    = (const float*)d_in[6];  // [100000 x 3 x 207]
  float* out = (float*)d_out;                        // [100000 x 3]

  float* ws       = (float*)d_ws;
  float* v_shaped = ws;                 // 300000
  float* v_posed  = ws + 300000;        // 300000
  float* Jws      = ws + 600000;        // 72
  float* lrot     = ws + 600072;        // 208
  float* Gws      = ws + 600280;        // 384

  const int blocksRows = (N_TILES_ROWS + 7) / 8;  // 2344 (8 waves / block)
  const int blocksVert = (N_TILES_VERT + 7) / 8;  // 782

  shape_blend_kernel<<<blocksRows, 256, 0, stream>>>(shapedirs, beta,
                                                     v_template, v_shaped);
  joint_reduce_kernel<<<72, 256, 0, stream>>>(J_regressor, v_shaped, Jws);
  pose_kernel<<<1, 32, 0, stream>>>(pose, Jws, lrot, Gws);
  pose_blend_kernel<<<blocksRows, 256, 0, stream>>>(posedirs, lrot,
                                                    v_shaped, v_posed);
  skin_kernel<<<blocksVert, 256, 0, stream>>>(skin_w, Gws, v_posed, out);
}